// WeightedKNNLoss_5016521802073
// MI455X (gfx1250) — compile-verified
//
#include <hip/hip_runtime.h>
#include <math.h>

typedef float v2f __attribute__((ext_vector_type(2)));
typedef float v8f __attribute__((ext_vector_type(8)));

#define EPSV 1e-5f
#define BOUNDARYV (1.0f - 1e-5f)
#define TILE 16
#define WAVES_PER_BLOCK 8
#define THREADS 256

// ---------------------------------------------------------------------------
// Kernel 1: per-row aux record: { sq, 1/(1-min(sq,BOUNDARY)), idx*N, idx }
// (one real division per row instead of one per matrix element)
// ---------------------------------------------------------------------------
__global__ void rowaux_kernel(const float* __restrict__ z,
                              const int* __restrict__ idx,
                              float4* __restrict__ aux, int B, int N) {
    int i = blockIdx.x * blockDim.x + threadIdx.x;
    if (i < B) {
        const float4* zr = (const float4*)(z + (size_t)i * 16);
        float s = 0.0f;
#pragma unroll
        for (int q = 0; q < 4; ++q) {
            float4 v = zr[q];
            s += v.x * v.x + v.y * v.y + v.z * v.z + v.w * v.w;
        }
        float inva = 1.0f / (1.0f - fminf(s, BOUNDARYV));
        int ind = idx[i];
        float4 a;
        a.x = s;
        a.y = inva;
        a.z = __int_as_float(ind * N);   // row offset into W (fits in 32 bits)
        a.w = __int_as_float(ind);
        aux[i] = a;
    }
}

// ---------------------------------------------------------------------------
// Kernel 2: one wave32 per 16x16 tile of the BxB matrix.
//   Gram tile via 4x V_WMMA_F32_16X16X4_F32 (K = 16 = 4*4), exact fp32.
//   Then Poincare distance + W gather + 4 weighted partial sums.
// ---------------------------------------------------------------------------
__global__ __launch_bounds__(THREADS) void poincare_tile_kernel(
    const float* __restrict__ z, const float* __restrict__ W,
    const float4* __restrict__ aux, float* __restrict__ partials,
    int tilesPerRow) {
    __shared__ float red[WAVES_PER_BLOCK][4];

    const int lane = threadIdx.x & 31;
    const int wave = threadIdx.x >> 5;
    const int tile = blockIdx.x * WAVES_PER_BLOCK + wave;
    const int ti = tile / tilesPerRow;
    const int tj = tile - ti * tilesPerRow;
    const int i0 = ti * TILE;
    const int j0 = tj * TILE;
    const int half = lane >> 4;   // half-wave: selects K pair (A/B), M offset (C)
    const int l    = lane & 15;   // row (A) / column (B,C) within tile

    // f32 16x4 A layout: lane l holds row (i0+l); VGPR0 = K0 (half0) / K2 (half1),
    // VGPR1 = K1 / K3.  B (4x16 = z.T columns) mirrors it with rows (j0+l).
    const float* arow = z + (size_t)(i0 + l) * 16 + 2 * half;
    const float* brow = z + (size_t)(j0 + l) * 16 + 2 * half;

    v8f c = {};  // C/D accumulator: c[r] = G[i0 + r + 8*half][j0 + l]
#pragma unroll
    for (int k = 0; k < 4; ++k) {     // K chunks of 4
        float2 av = *(const float2*)(arow + 4 * k);
        float2 bv = *(const float2*)(brow + 4 * k);
        v2f a, b;
        a[0] = av.x; a[1] = av.y;
        b[0] = bv.x; b[1] = bv.y;
        // 8 args: (neg_a, A, neg_b, B, c_mod, C, reuse_a, reuse_b)
        c = __builtin_amdgcn_wmma_f32_16x16x4_f32(
            false, a, false, b, (short)0, c, false, false);
    }

    // Column-side (j) quantities: one per lane.
    const int      j    = j0 + l;
    const float4   auxj = aux[j];
    const float    sqj  = auxj.x;
    const float    invaj = auxj.y;
    const unsigned indj  = (unsigned)__float_as_int(auxj.w);

    float an = 0.0f, ad = 0.0f, rn = 0.0f, rd = 0.0f;
#pragma unroll
    for (int r = 0; r < 8; ++r) {
        const int      i     = i0 + r + 8 * half;
        const float4   auxi  = aux[i];          // uniform across 16 lanes -> L0 hit
        const float    sqi   = auxi.x;
        const float    invai = auxi.y;
        const unsigned woff  = (unsigned)__float_as_int(auxi.z) + indj;

        const float g      = c[r];
        const float sqdist = fmaxf(sqi + sqj - 2.0f * g, 0.0f);
        // x = 2*sqdist/(alpha_i*alpha_j) + 1  via precomputed reciprocals
        const float x  = fmaf(2.0f * sqdist, invai * invaj, 1.0f);
        const float t  = __builtin_amdgcn_sqrtf(fmaxf(x * x - 1.0f, EPSV * EPSV));
        const float xt = x + t;
        const float dlow = __logf(xt);          // v_log_f32

        float w = W[woff];                      // 32-bit gather offset
        const bool diag = (i == j);
        if (diag) w = 0.0f;
        const float nn = (diag || (w != 0.0f)) ? 0.0f : 1.0f;

        an += w * dlow;
        ad += w;
        rn += nn * __builtin_amdgcn_rcpf(xt);   // exp(-log(xt)) == 1/xt
        rd += nn;
    }

    // wave32 butterfly reduction
#pragma unroll
    for (int off = 16; off > 0; off >>= 1) {
        an += __shfl_xor(an, off, 32);
        ad += __shfl_xor(ad, off, 32);
        rn += __shfl_xor(rn, off, 32);
        rd += __shfl_xor(rd, off, 32);
    }
    if (lane == 0) {
        red[wave][0] = an; red[wave][1] = ad;
        red[wave][2] = rn; red[wave][3] = rd;
    }
    __syncthreads();
    if (threadIdx.x < 4) {
        float s = 0.0f;
#pragma unroll
        for (int wv = 0; wv < WAVES_PER_BLOCK; ++wv) s += red[wv][threadIdx.x];
        partials[(size_t)blockIdx.x * 4 + threadIdx.x] = s;  // overwrite: deterministic
    }
}

// ---------------------------------------------------------------------------
// Kernel 3: reduce per-block partials, emit scalar loss.
// ---------------------------------------------------------------------------
__global__ __launch_bounds__(256) void finalize_kernel(
    const float* __restrict__ partials, int nblocks, float* __restrict__ out) {
    __shared__ float sm[4][256];
    const int t = threadIdx.x;
    float a0 = 0.0f, a1 = 0.0f, a2 = 0.0f, a3 = 0.0f;
    for (int b = t; b < nblocks; b += 256) {
        a0 += partials[(size_t)b * 4 + 0];
        a1 += partials[(size_t)b * 4 + 1];
        a2 += partials[(size_t)b * 4 + 2];
        a3 += partials[(size_t)b * 4 + 3];
    }
    sm[0][t] = a0; sm[1][t] = a1; sm[2][t] = a2; sm[3][t] = a3;
    __syncthreads();
    for (int off = 128; off > 0; off >>= 1) {
        if (t < off) {
            sm[0][t] += sm[0][t + off];
            sm[1][t] += sm[1][t + off];
            sm[2][t] += sm[2][t + off];
            sm[3][t] += sm[3][t + off];
        }
        __syncthreads();
    }
    if (t == 0) {
        out[0] = sm[0][0] / (sm[1][0] + 1e-8f) +
                 sm[2][0] / (sm[3][0] + 1e-8f);
    }
}

// ---------------------------------------------------------------------------
extern "C" void kernel_launch(void* const* d_in, const int* in_sizes, int n_in,
                              void* d_out, int out_size, void* d_ws, size_t ws_size,
                              hipStream_t stream) {
    const float* z   = (const float*)d_in[0];   // [B,16]
    const float* W   = (const float*)d_in[1];   // [N,N]
    const int*   idx = (const int*)d_in[2];     // [B]
    float*       out = (float*)d_out;           // scalar

    const int B = in_sizes[2];
    const int N = (int)(sqrt((double)in_sizes[1]) + 0.5);

    float4* aux      = (float4*)d_ws;               // B float4 records
    float*  partials = (float*)d_ws + 4 * (size_t)B;

    const int tilesPerRow = B / TILE;
    const int ntiles      = tilesPerRow * tilesPerRow;
    const int nblocks     = ntiles / WAVES_PER_BLOCK;

    rowaux_kernel<<<(B + 255) / 256, 256, 0, stream>>>(z, idx, aux, B, N);
    poincare_tile_kernel<<<nblocks, THREADS, 0, stream>>>(
        z, W, aux, partials, tilesPerRow);
    finalize_kernel<<<1, 256, 0, stream>>>(partials, nblocks, out);
}